// V1EdgeTimeMixer_2791728742596
// MI455X (gfx1250) — compile-verified
//
#include <hip/hip_runtime.h>
#include <hip/hip_bf16.h>

// ---------------------------------------------------------------------------
// Fused V1EdgeTimeMixer for MI455X (gfx1250, wave32, WMMA).
// One workgroup per (b, f); streams T in 32-column chunks fully in LDS.
// GEMMs: V_WMMA_F32_16X16X4_F32, one K-pass per wave with all accumulators
// live (each B fragment read from LDS exactly once per wave).
// X tiles: double-buffered GLOBAL_LOAD_ASYNC_TO_LDS_B32 prefetch (ASYNCcnt),
// overlapping the next chunk's HBM/L2 traffic with current-chunk WMMAs.
// ---------------------------------------------------------------------------

typedef __attribute__((ext_vector_type(2))) float v2f;
typedef __attribute__((ext_vector_type(8))) float v8f;

namespace cfg {
constexpr int B = 8;
constexpr int C = 256;   // input / output channels
constexpr int F = 64;
constexpr int T = 500;
constexpr int H = 512;   // hidden channels
constexpr int SW = 2;    // carried state width (K-1)
constexpr int TC = 32;   // time-columns per chunk
constexpr int NCHUNK = (T + TC - 1) / TC;  // 16 (last chunk: 20 valid cols)
constexpr int NTHREADS = 256;              // 8 wave32s
// LDS strides (floats). Transposed tiles use stride = K + 4 so that
// stride % 64 == 4 -> b64 B-fragment reads are bank-conflict-free.
constexpr int SX  = C + 4;   // 260, XsT  [2][TC][SX]
constexpr int SH  = H + 4;   // 516, H2sT [TC][SH]
constexpr int SH1 = TC + 2;  // 34,  H1s  [H][SH1] (row-major)
constexpr int SO  = TC + 2;  // 34,  O3s  [C][SO]  (row-major)
}

// Column-wise stats over ROWS rows, 32 cols; S row-major with STRIDE.
template <int ROWS, int STRIDE>
__device__ __forceinline__ void col_stats(const float* S, float* redS, float* redQ,
                                          float* statM, float* statR, int tid) {
  const int j = tid & 31;
  const int p = tid >> 5;
  constexpr int RP = ROWS / 8;
  float s = 0.f, q = 0.f;
#pragma unroll 4
  for (int i = 0; i < RP; ++i) {
    float v = S[(p * RP + i) * STRIDE + j];
    s += v;
    q += v * v;
  }
  redS[p * 32 + j] = s;
  redQ[p * 32 + j] = q;
  __syncthreads();
  if (tid < 32) {
    float s2 = 0.f, q2 = 0.f;
#pragma unroll
    for (int pp = 0; pp < 8; ++pp) {
      s2 += redS[pp * 32 + tid];
      q2 += redQ[pp * 32 + tid];
    }
    float mean = s2 * (1.0f / ROWS);
    float var = fmaf(-mean, mean, q2 * (1.0f / ROWS));
    statM[tid] = mean;
    statR[tid] = rsqrtf(var + 1e-8f);
  }
  __syncthreads();
}

template <int ROWS, int STRIDE>
__device__ __forceinline__ void norm_apply(float* S, const float* __restrict__ g,
                                           const float* __restrict__ bt,
                                           const float* statM, const float* statR,
                                           int tid) {
  const int j = tid & 31;
  const int p = tid >> 5;
  constexpr int RP = ROWS / 8;
  float mean = statM[j], inv = statR[j];
#pragma unroll 4
  for (int i = 0; i < RP; ++i) {
    int r = p * RP + i;
    float v = S[r * STRIDE + j];
    S[r * STRIDE + j] = fmaf((v - mean) * inv, g[r], bt[r]);
  }
  __syncthreads();
}

// Transposed variants: S is [32 cols][STRIDE], stats over ROWS per column.
template <int ROWS, int STRIDE>
__device__ __forceinline__ void col_stats_T(const float* S, float* redS, float* redQ,
                                            float* statM, float* statR, int tid) {
  const int j = tid & 31;
  const int p = tid >> 5;
  constexpr int RP = ROWS / 8;
  const float* base = S + j * STRIDE + p * RP;
  float s = 0.f, q = 0.f;
#pragma unroll 4
  for (int i = 0; i < RP; ++i) {
    float v = base[i];
    s += v;
    q += v * v;
  }
  redS[p * 32 + j] = s;
  redQ[p * 32 + j] = q;
  __syncthreads();
  if (tid < 32) {
    float s2 = 0.f, q2 = 0.f;
#pragma unroll
    for (int pp = 0; pp < 8; ++pp) {
      s2 += redS[pp * 32 + tid];
      q2 += redQ[pp * 32 + tid];
    }
    float mean = s2 * (1.0f / ROWS);
    float var = fmaf(-mean, mean, q2 * (1.0f / ROWS));
    statM[tid] = mean;
    statR[tid] = rsqrtf(var + 1e-8f);
  }
  __syncthreads();
}

template <int ROWS, int STRIDE>
__device__ __forceinline__ void norm_apply_T(float* S, const float* __restrict__ g,
                                             const float* __restrict__ bt,
                                             const float* statM, const float* statR,
                                             int tid) {
  const int j = tid & 31;
  const int p = tid >> 5;
  constexpr int RP = ROWS / 8;
  float mean = statM[j], inv = statR[j];
  float* base = S + j * STRIDE + p * RP;
#pragma unroll 4
  for (int i = 0; i < RP; ++i) {
    int r = p * RP + i;
    float v = base[i];
    base[i] = fmaf((v - mean) * inv, g[r], bt[r]);
  }
  __syncthreads();
}

// Issue per-lane async DMA of one full X chunk (32 t-columns, 256 channels)
// into the transposed LDS tile at lds_base. Tracked by ASYNCcnt.
__device__ __forceinline__ void issue_x_async(uint32_t lds_base, uint64_t xb_addr,
                                              int t0, int tid) {
  const int j = tid & 31;       // time column (coalesced along t)
  const int c0 = tid >> 5;
#pragma unroll 4
  for (int c = c0; c < cfg::C; c += 8) {
    uint32_t voff = (uint32_t)(((uint32_t)c * (uint32_t)(cfg::F * cfg::T) +
                                (uint32_t)(t0 + j)) * 4u);
    uint32_t dst = lds_base + (uint32_t)((j * cfg::SX + c) * 4);
    asm volatile("global_load_async_to_lds_b32 %0, %1, %2"
                 :
                 : "v"(dst), "v"(voff), "s"(xb_addr)
                 : "memory");
  }
}

__global__ __launch_bounds__(cfg::NTHREADS) void v1edge_mixer_kernel(
    const float* __restrict__ x, const float* __restrict__ state_in,
    const float* __restrict__ pre_w, const float* __restrict__ pre_b,
    const float* __restrict__ pre_nw, const float* __restrict__ pre_nb,
    const float* __restrict__ dw_w, const float* __restrict__ dw_b,
    const float* __restrict__ dw_nw, const float* __restrict__ dw_nb,
    const float* __restrict__ post_w, const float* __restrict__ post_b,
    const float* __restrict__ post_nw, const float* __restrict__ post_nb,
    float* __restrict__ out) {
  using namespace cfg;

  // ~237 KB LDS: fits the 320 KB/WGP budget of gfx1250 (one WG per WGP).
  __shared__ float XsT[2][TC * SX]; //  65 KB  input tiles, transposed, dbl-buf
  __shared__ float H1s[H * SH1];    //  68 KB  pre-GEMM out, row-major [h][t]
  __shared__ float H2sT[TC * SH];   //  64.5KB depthwise out, transposed [t][h]
  __shared__ float O3s[C * SO];     //  34 KB  post-GEMM out, row-major [c][t]
  __shared__ float carry[H * SW];   //   4 KB  last 2 normalized columns
  __shared__ float redS[8 * 32];
  __shared__ float redQ[8 * 32];
  __shared__ float statM[32];
  __shared__ float statR[32];

  const int tid = threadIdx.x;
  const int lane = tid & 31;
  const int wave = tid >> 5;
  const int b = blockIdx.x / F;
  const int f = blockIdx.x % F;

  const float* xb = x + (size_t)b * C * F * T + (size_t)f * T;  // + c*F*T + t
  float* yb = out + (size_t)b * C * F * T + (size_t)f * T;
  float* state_out = out + (size_t)B * C * F * T;

  const uint64_t xb_addr = (uint64_t)(uintptr_t)xb;              // uniform SADDR
  const uint32_t xs_lds0 = (uint32_t)(uintptr_t)(&XsT[0][0]);    // LDS offsets
  const uint32_t xs_lds1 = (uint32_t)(uintptr_t)(&XsT[1][0]);

  // WMMA lane geometry
  const int lrow = lane & 15;     // A/B row-within-tile, D column lane
  const int lhalf = (lane >> 4);  // half-wave select
  const int kofs = 2 * lhalf;     // K offset of fragment element 0

  // Load incoming depthwise state: state[b, h, f, s]
  for (int i = tid; i < H * SW; i += NTHREADS) {
    int r = i >> 1, sidx = i & 1;
    carry[i] = state_in[(((size_t)b * H + r) * F + f) * SW + sidx];
  }

  // Kick off DMA of chunk 0.
  issue_x_async(xs_lds0, xb_addr, 0, tid);

  for (int ch = 0; ch < NCHUNK; ++ch) {
    const int t0 = ch * TC;
    const int nvalid = (T - t0 < TC) ? (T - t0) : TC;
    float* Xcur = &XsT[ch & 1][0];

    if (nvalid == TC) {
      // Drain this chunk's DMA, then immediately prefetch the next full chunk
      // into the other buffer; its latency hides behind this chunk's WMMAs.
      asm volatile("s_wait_asynccnt 0x0" ::: "memory");
      __syncthreads();
      if (ch + 1 < NCHUNK - 1) {
        issue_x_async((ch & 1) ? xs_lds0 : xs_lds1, xb_addr, t0 + TC, tid);
      }
    } else {
      // Ragged tail: scalar path with zero padding.
      const int j = tid & 31;
      const int c0 = tid >> 5;
      const int t = t0 + j;
#pragma unroll 4
      for (int c = c0; c < C; c += 8) {
        Xcur[j * SX + c] = (t < T) ? xb[(size_t)c * F * T + t] : 0.0f;
      }
      __syncthreads();
    }

    // ---- Stage 1: H1 = relu(pre_w(512x256) @ X(256x32) + pre_b) ----
    // One K-pass per wave: 4 M-tiles x 2 N-tiles, 8 live accumulators;
    // each B fragment is read from LDS exactly once.
    {
      const float* aptr = pre_w + (size_t)(wave * 16 + lrow) * C + kofs;
      const float* bp0 = &Xcur[lrow * SX + kofs];
      const float* bp1 = &Xcur[(lrow + 16) * SX + kofs];
      v8f acc[8] = {};
#pragma unroll 2
      for (int ks = 0; ks < C / 4; ++ks) {
        v2f b0 = *(const v2f*)(bp0 + ks * 4);  // ds_load_b64, conflict-free
        v2f b1 = *(const v2f*)(bp1 + ks * 4);
#pragma unroll
        for (int m = 0; m < 4; ++m) {
          v2f a = *(const v2f*)(aptr + (size_t)m * 128 * C + ks * 4);
          acc[2 * m] = __builtin_amdgcn_wmma_f32_16x16x4_f32(
              false, a, false, b0, (short)0, acc[2 * m], false, false);
          acc[2 * m + 1] = __builtin_amdgcn_wmma_f32_16x16x4_f32(
              false, a, false, b1, (short)0, acc[2 * m + 1], false, false);
        }
      }
#pragma unroll
      for (int m = 0; m < 4; ++m) {
        const int rbase = (wave + 8 * m) * 16 + 8 * lhalf;
#pragma unroll
        for (int v = 0; v < 8; ++v) {
          int r = rbase + v;
          float bias = pre_b[r];
          H1s[r * SH1 + lrow] = fmaxf(acc[2 * m][v] + bias, 0.0f);
          H1s[r * SH1 + 16 + lrow] = fmaxf(acc[2 * m + 1][v] + bias, 0.0f);
        }
      }
    }
    __syncthreads();

    // ---- channel_norm over H=512 per column (H1, row-major) ----
    col_stats<H, SH1>(H1s, redS, redQ, statM, statR, tid);
    norm_apply<H, SH1>(H1s, pre_nw, pre_nb, statM, statR, tid);

    // ---- Depthwise causal conv (K=3) along t + relu -> H2sT ----
    {
      const int j = tid & 31;
      const int p = tid >> 5;
#pragma unroll 4
      for (int i = 0; i < H / 8; ++i) {
        int r = p * (H / 8) + i;
        float w0 = dw_w[r * 3 + 0];
        float w1 = dw_w[r * 3 + 1];
        float w2 = dw_w[r * 3 + 2];
        float xm2 = (j >= 2) ? H1s[r * SH1 + j - 2] : carry[r * SW + j];
        float xm1 = (j >= 1) ? H1s[r * SH1 + j - 1] : carry[r * SW + 1];
        float xc = H1s[r * SH1 + j];
        float o = fmaf(w0, xm2, fmaf(w1, xm1, w2 * xc)) + dw_b[r];
        H2sT[j * SH + r] = fmaxf(o, 0.0f);
      }
    }
    __syncthreads();

    // ---- Update carry with the last 2 *valid* normalized columns ----
    for (int i = tid; i < H * SW; i += NTHREADS) {
      int r = i >> 1, sidx = i & 1;
      carry[i] = H1s[r * SH1 + (nvalid - SW + sidx)];
    }
    // (carry/H1s disjoint from H2sT/red/stat used next; barriers follow)

    // ---- channel_norm over H=512 on H2 (transposed layout) ----
    col_stats_T<H, SH>(H2sT, redS, redQ, statM, statR, tid);
    norm_apply_T<H, SH>(H2sT, dw_nw, dw_nb, statM, statR, tid);

    // ---- Stage 3: O3 = post_w(256x512) @ H2(512x32) + post_b ----
    // One K-pass per wave: 2 M-tiles x 2 N-tiles, 4 live accumulators.
    {
      const float* aptr = post_w + (size_t)(wave * 16 + lrow) * H + kofs;
      const float* bp0 = &H2sT[lrow * SH + kofs];
      const float* bp1 = &H2sT[(lrow + 16) * SH + kofs];
      v8f acc[4] = {};
#pragma unroll 2
      for (int ks = 0; ks < H / 4; ++ks) {
        v2f b0 = *(const v2f*)(bp0 + ks * 4);
        v2f b1 = *(const v2f*)(bp1 + ks * 4);
#pragma unroll
        for (int m = 0; m < 2; ++m) {
          v2f a = *(const v2f*)(aptr + (size_t)m * 128 * H + ks * 4);
          acc[2 * m] = __builtin_amdgcn_wmma_f32_16x16x4_f32(
              false, a, false, b0, (short)0, acc[2 * m], false, false);
          acc[2 * m + 1] = __builtin_amdgcn_wmma_f32_16x16x4_f32(
              false, a, false, b1, (short)0, acc[2 * m + 1], false, false);
        }
      }
#pragma unroll
      for (int m = 0; m < 2; ++m) {
        const int rbase = (wave + 8 * m) * 16 + 8 * lhalf;
#pragma unroll
        for (int v = 0; v < 8; ++v) {
          int r = rbase + v;
          float bias = post_b[r];
          O3s[r * SO + lrow] = acc[2 * m][v] + bias;
          O3s[r * SO + 16 + lrow] = acc[2 * m + 1][v] + bias;
        }
      }
    }
    __syncthreads();

    // ---- channel_norm over C=256 + residual + store ----
    col_stats<C, SO>(O3s, redS, redQ, statM, statR, tid);
    {
      const int j = tid & 31;
      const int p = tid >> 5;
      const float mean = statM[j], inv = statR[j];
#pragma unroll 4
      for (int i = 0; i < C / 8; ++i) {
        int r = p * (C / 8) + i;
        float v = O3s[r * SO + j];
        float yv = fmaf((v - mean) * inv, post_nw[r], post_nb[r]) +
                   Xcur[j * SX + r];
        if (j < nvalid) {
          yb[(size_t)r * F * T + t0 + j] = yv;
        }
      }
    }
    __syncthreads();  // protect X/O3s/stat before next chunk
  }

  // ---- Emit new_state = last 2 normalized pre-branch columns (t=498,499) ----
  for (int i = tid; i < H * SW; i += NTHREADS) {
    int r = i >> 1, sidx = i & 1;
    state_out[(((size_t)b * H + r) * F + f) * SW + sidx] = carry[i];
  }
}

extern "C" void kernel_launch(void* const* d_in, const int* in_sizes, int n_in,
                              void* d_out, int out_size, void* d_ws, size_t ws_size,
                              hipStream_t stream) {
  (void)in_sizes; (void)n_in; (void)out_size; (void)d_ws; (void)ws_size;
  const float* x       = (const float*)d_in[0];
  const float* state   = (const float*)d_in[1];
  const float* pre_w   = (const float*)d_in[2];
  const float* pre_b   = (const float*)d_in[3];
  const float* pre_nw  = (const float*)d_in[4];
  const float* pre_nb  = (const float*)d_in[5];
  const float* dw_w    = (const float*)d_in[6];
  const float* dw_b    = (const float*)d_in[7];
  const float* dw_nw   = (const float*)d_in[8];
  const float* dw_nb   = (const float*)d_in[9];
  const float* post_w  = (const float*)d_in[10];
  const float* post_b  = (const float*)d_in[11];
  const float* post_nw = (const float*)d_in[12];
  const float* post_nb = (const float*)d_in[13];
  float* out = (float*)d_out;

  dim3 grid(cfg::B * cfg::F);  // 512 workgroups, one per (b, f)
  dim3 block(cfg::NTHREADS);   // 8 wave32s
  v1edge_mixer_kernel<<<grid, block, 0, stream>>>(
      x, state, pre_w, pre_b, pre_nw, pre_nb, dw_w, dw_b, dw_nw, dw_nb,
      post_w, post_b, post_nw, post_nb, out);
}